// ChamferDistance_30124900614424
// MI455X (gfx1250) — compile-verified
//
#include <hip/hip_runtime.h>

typedef __attribute__((ext_vector_type(2))) float v2f;
typedef __attribute__((ext_vector_type(8))) float v8f;

// One wave (32 lanes) handles one 16-point "query" tile of X for one batch,
// scanning all of Y to find, for each query point, min_j ||x_i - y_j||^2.
//
// WMMA V_WMMA_F32_16X16X4_F32 operand layout (wave32):
//   A (16x4):  lanes 0-15: VGPR0=K0, VGPR1=K1 for row M=lane
//              lanes 16-31: VGPR0=K2, VGPR1=K3 for row M=lane-16
//   B (4x16):  lanes 0-15: VGPR0=K0 row, VGPR1=K1 row, col N=lane
//              lanes 16-31: VGPR0=K2, VGPR1=K3, col N=lane-16
//   C/D (16x16): VGPR r: lanes 0-15 -> M=r, lanes 16-31 -> M=r+8; N=lane&15
//
// We pre-scale A by -2 so the accumulator directly holds -2 * x.y, and track
// min over columns of (||y||^2 - 2 x.y); ||x||^2 is added once at the end.
__global__ __launch_bounds__(256) void chamfer_min_kernel(
    const float* __restrict__ X,   // (B, nx, 3) query set
    const float* __restrict__ Y,   // (B, ny, 3) reference set
    int nx, int ny,
    float* __restrict__ outmin)    // (B, nx) per-query-point min sq dist
{
    const int lane = threadIdx.x & 31;
    const int wave = threadIdx.x >> 5;
    const int gw   = blockIdx.x * (blockDim.x >> 5) + wave;   // global wave id
    const int qtiles = nx >> 4;
    const int b = gw / qtiles;
    const int q = gw % qtiles;

    const float* Xb = X + (size_t)b * nx * 3;
    const float* Yb = Y + (size_t)b * ny * 3;

    const int  row = lane & 15;
    const bool hi  = lane >= 16;

    // Load this lane's query point (rows duplicated across the two halves).
    const float* xp = Xb + (size_t)(q * 16 + row) * 3;
    const float xx = xp[0], xy = xp[1], xz = xp[2];
    const float x2 = xx * xx + xy * xy + xz * xz;   // ||x_row||^2, per lane

    // A operand, pre-scaled by -2; K=3 padded to 4 with zero.
    v2f a;
    a.x = hi ? (-2.0f * xz) : (-2.0f * xx);
    a.y = hi ? 0.0f         : (-2.0f * xy);

    v8f mins;
#pragma unroll
    for (int r = 0; r < 8; ++r) mins[r] = 3.0e38f;

    const int ttiles = ny >> 4;
    for (int t = 0; t < ttiles; ++t) {
        // Load target point for this lane's column (both halves load same 16 pts).
        const float* yp = Yb + (size_t)(t * 16 + row) * 3;
        const float yx = yp[0], yy = yp[1], yz = yp[2];
        const float y2 = yx * yx + yy * yy + yz * yz;  // ||y_col||^2, per lane

        v2f bb;
        bb.x = hi ? yz   : yx;
        bb.y = hi ? 0.0f : yy;

        v8f c = {};
        // 8 args: (neg_a, A, neg_b, B, c_mod, C, reuse_a, reuse_b)
        c = __builtin_amdgcn_wmma_f32_16x16x4_f32(
                /*neg_a=*/false, a, /*neg_b=*/false, bb,
                /*c_mod=*/(short)0, c, /*reuse_a=*/false, /*reuse_b=*/false);

#pragma unroll
        for (int r = 0; r < 8; ++r) {
            // c[r] = -2 * x_row . y_col ; candidate = ||y||^2 - 2 x.y
            mins[r] = fminf(mins[r], y2 + c[r]);
        }
    }

    // Min-reduce across the 16 columns (lanes) within each 16-lane half.
#pragma unroll
    for (int m = 1; m < 16; m <<= 1) {
#pragma unroll
        for (int r = 0; r < 8; ++r) {
            mins[r] = fminf(mins[r], __shfl_xor(mins[r], m, 32));
        }
    }

    // Add ||x_row||^2 per row. After reduction:
    //   lanes 0-15  all hold rows 0..7  in regs 0..7
    //   lanes 16-31 all hold rows 8..15 in regs 0..7
    // x2 for row r lives in lanes r and 16+r; fetch with bpermute while all
    // lanes are still active, then lane 0 / lane 16 write out 8 rows each.
    float val[8];
#pragma unroll
    for (int r = 0; r < 8; ++r) {
        const int src = hi ? (8 + r) : r;          // x2 of row (hi ? 8+r : r)
        val[r] = mins[r] + __shfl(x2, src, 32);
    }

    if (lane == 0 || lane == 16) {
        float* dst = outmin + (size_t)b * nx + q * 16 + (hi ? 8 : 0);
#pragma unroll
        for (int r = 0; r < 8; ++r) dst[r] = val[r];
    }
}

// Single-block final reduction:
// loss = (1/(B*N)) sum smin + (1/(B*M)) sum tmin
__global__ __launch_bounds__(256) void chamfer_reduce_kernel(
    const float* __restrict__ smin, const float* __restrict__ tmin,
    int n_s, int n_t, float inv_s, float inv_t, float* __restrict__ out)
{
    __shared__ float sdata[256];
    float acc = 0.0f;
    for (int i = threadIdx.x; i < n_s; i += 256) acc += smin[i] * inv_s;
    for (int i = threadIdx.x; i < n_t; i += 256) acc += tmin[i] * inv_t;
    sdata[threadIdx.x] = acc;
    __syncthreads();
    for (int s = 128; s > 0; s >>= 1) {
        if (threadIdx.x < (unsigned)s) sdata[threadIdx.x] += sdata[threadIdx.x + s];
        __syncthreads();
    }
    if (threadIdx.x == 0) out[0] = sdata[0];
}

extern "C" void kernel_launch(void* const* d_in, const int* in_sizes, int n_in,
                              void* d_out, int out_size, void* d_ws, size_t ws_size,
                              hipStream_t stream) {
    const float* src = (const float*)d_in[0];   // (B, N, 3) f32
    const float* tgt = (const float*)d_in[1];   // (B, M, 3) f32
    const int B = 4;
    const int N = in_sizes[0] / (B * 3);
    const int M = in_sizes[1] / (B * 3);

    float* smin = (float*)d_ws;                 // B*N floats
    float* tmin = smin + (size_t)B * N;         // B*M floats

    const int wavesPerBlock = 8;                // 256 threads
    const int gridS = (B * (N / 16)) / wavesPerBlock;   // 128 blocks
    const int gridT = (B * (M / 16)) / wavesPerBlock;

    // source -> target direction: per-source-point min over targets
    chamfer_min_kernel<<<gridS, 256, 0, stream>>>(src, tgt, N, M, smin);
    // target -> source direction: per-target-point min over sources
    chamfer_min_kernel<<<gridT, 256, 0, stream>>>(tgt, src, M, N, tmin);

    chamfer_reduce_kernel<<<1, 256, 0, stream>>>(
        smin, tmin, B * N, B * M, 1.0f / (B * N), 1.0f / (B * M), (float*)d_out);
}